// GATModule_64579128263109
// MI455X (gfx1250) — compile-verified
//
#include <hip/hip_runtime.h>
#include <hip/hip_bf16.h>
#include <math.h>

// ---------------- constants (match reference) ----------------
#define HID      128
#define HEADS    8
#define HEAD_DIM 16
#define LAYERS   3
#define NEG_SLOPE 0.2f
#define EPS_F     1e-16f

typedef float v2f __attribute__((ext_vector_type(2)));
typedef float v8f __attribute__((ext_vector_type(8)));

// ---------------- fp32 WMMA dual GEMM: XL = X@Wl, XR = X@Wr ----------------
// One block = 16 rows of X. 8 waves, wave w computes the 16x16 tile at
// columns [16w, 16w+16). K looped in steps of 4 -> v_wmma_f32_16x16x4_f32.
__global__ __launch_bounds__(256)
void gemm_dual_f32(const float* __restrict__ X,
                   const float* __restrict__ Wl,
                   const float* __restrict__ Wr,
                   float* __restrict__ XL,
                   float* __restrict__ XR) {
    __shared__ float As[16][HID];             // 8 KB A strip
    const int row0 = blockIdx.x * 16;

    // cooperative load of 16x128 A strip (512 float4s, 2 per thread)
    {
        const float4* gx = (const float4*)(X + (size_t)row0 * HID);
        float4* sx = (float4*)(&As[0][0]);
        for (int t = threadIdx.x; t < 16 * HID / 4; t += 256) sx[t] = gx[t];
    }
    __syncthreads();

    const int wave = threadIdx.x >> 5;        // 0..7 -> column tile
    const int lane = threadIdx.x & 31;
    const int col0 = wave * 16;
    const int half = lane >> 4;               // 0 or 1
    const int lm   = lane & 15;               // M for A, N for B/C/D

    v8f accL = {0.f,0.f,0.f,0.f,0.f,0.f,0.f,0.f};
    v8f accR = {0.f,0.f,0.f,0.f,0.f,0.f,0.f,0.f};

    #pragma unroll 4
    for (int k0 = 0; k0 < HID; k0 += 4) {
        const int ka = k0 + 2 * half;
        v2f a;                                 // A(M=lm, K=ka..ka+1)
        a.x = As[lm][ka];
        a.y = As[lm][ka + 1];

        v2f bl, br;                            // B(K=ka..ka+1, N=col0+lm)
        bl.x = Wl[(size_t)(ka + 0) * HID + col0 + lm];
        bl.y = Wl[(size_t)(ka + 1) * HID + col0 + lm];
        br.x = Wr[(size_t)(ka + 0) * HID + col0 + lm];
        br.y = Wr[(size_t)(ka + 1) * HID + col0 + lm];

        accL = __builtin_amdgcn_wmma_f32_16x16x4_f32(false, a, false, bl,
                                                     (short)0, accL, false, false);
        accR = __builtin_amdgcn_wmma_f32_16x16x4_f32(false, a, false, br,
                                                     (short)0, accR, false, false);
    }

    // C/D layout: lanes 0-15 -> M = r, lanes 16-31 -> M = 8 + r; N = lm
    #pragma unroll
    for (int r = 0; r < 8; ++r) {
        const int m = r + 8 * half;
        XL[(size_t)(row0 + m) * HID + col0 + lm] = accL[r];
        XR[(size_t)(row0 + m) * HID + col0 + lm] = accR[r];
    }
}

// ---------------- helpers ----------------
__device__ __forceinline__ void atomicMaxFloat(float* addr, float val) {
    // valid when the location is initialized to -inf and accessed only via this op
    if (val >= 0.0f)
        atomicMax((int*)addr, __float_as_int(val));
    else
        atomicMin((unsigned int*)addr, __float_as_uint(val));
}

__global__ void init_md_kernel(float* __restrict__ mmax,
                               float* __restrict__ denom, int n) {
    int i = blockIdx.x * blockDim.x + threadIdx.x;
    if (i < n) { mmax[i] = -INFINITY; denom[i] = 0.0f; }
}

__global__ void init_out_kernel(float* __restrict__ out,
                                const float* __restrict__ bias, int total) {
    int i = blockIdx.x * blockDim.x + threadIdx.x;
    if (i < total) out[i] = bias[i & (HID - 1)];
}

// ---------------- edge stage ----------------
// logits[e,h] = sum_d att[h,d] * leaky_relu(xl[src,h,d] + xr[dst,h,d]);
// also scatter-max into mmax[dst,h].
__global__ void edge_logits_kernel(const float* __restrict__ xl,
                                   const float* __restrict__ xr,
                                   const int* __restrict__ src,
                                   const int* __restrict__ dst,
                                   const float* __restrict__ att,
                                   float* __restrict__ logits,
                                   float* __restrict__ mmax, int EH) {
    int i = blockIdx.x * blockDim.x + threadIdx.x;
    if (i >= EH) return;
    const int e = i / HEADS, h = i % HEADS;
    const int s = src[e], d = dst[e];

    const float4* pl = (const float4*)(xl + (size_t)s * HID + h * HEAD_DIM);
    const float4* pr = (const float4*)(xr + (size_t)d * HID + h * HEAD_DIM);
    const float4* pa = (const float4*)(att + h * HEAD_DIM);

    float acc = 0.0f;
    #pragma unroll
    for (int q = 0; q < 4; ++q) {
        float4 a = pl[q], b = pr[q], w = pa[q];
        float v;
        v = a.x + b.x; v = (v > 0.0f) ? v : NEG_SLOPE * v; acc += v * w.x;
        v = a.y + b.y; v = (v > 0.0f) ? v : NEG_SLOPE * v; acc += v * w.y;
        v = a.z + b.z; v = (v > 0.0f) ? v : NEG_SLOPE * v; acc += v * w.z;
        v = a.w + b.w; v = (v > 0.0f) ? v : NEG_SLOPE * v; acc += v * w.w;
    }
    logits[i] = acc;
    atomicMaxFloat(&mmax[(size_t)d * HEADS + h], acc);
}

// w = exp(logit - mmax[dst,h]) stored in place; denom[dst,h] += w
__global__ void edge_softmax_kernel(const int* __restrict__ dst,
                                    const float* __restrict__ mmax,
                                    float* __restrict__ lw,
                                    float* __restrict__ denom, int EH) {
    int i = blockIdx.x * blockDim.x + threadIdx.x;
    if (i >= EH) return;
    const int e = i / HEADS, h = i % HEADS;
    const int d = dst[e];
    const float w = expf(lw[i] - mmax[(size_t)d * HEADS + h]);
    lw[i] = w;
    atomicAdd(&denom[(size_t)d * HEADS + h], w);
}

// out[dst, h*16+d] += (w / (denom+eps)) * xl[src, h*16+d]
__global__ void edge_aggregate_kernel(const int* __restrict__ src,
                                      const int* __restrict__ dst,
                                      const float* __restrict__ xl,
                                      const float* __restrict__ lw,
                                      const float* __restrict__ denom,
                                      float* __restrict__ out, int EH) {
    int i = blockIdx.x * blockDim.x + threadIdx.x;
    if (i >= EH) return;
    const int e = i / HEADS, h = i % HEADS;
    const int s = src[e], d = dst[e];
    const float alpha = lw[i] / (denom[(size_t)d * HEADS + h] + EPS_F);

    const float4* pl = (const float4*)(xl + (size_t)s * HID + h * HEAD_DIM);
    float* po = out + (size_t)d * HID + h * HEAD_DIM;
    #pragma unroll
    for (int q = 0; q < 4; ++q) {
        float4 v = pl[q];
        atomicAdd(po + 4 * q + 0, alpha * v.x);
        atomicAdd(po + 4 * q + 1, alpha * v.y);
        atomicAdd(po + 4 * q + 2, alpha * v.z);
        atomicAdd(po + 4 * q + 3, alpha * v.w);
    }
}

// ---------------- host-side orchestration ----------------
extern "C" void kernel_launch(void* const* d_in, const int* in_sizes, int n_in,
                              void* d_out, int out_size, void* d_ws, size_t ws_size,
                              hipStream_t stream) {
    const float* x          = (const float*)d_in[0];
    const int*   edge_index = (const int*)  d_in[1];   // [2, E]
    const float* Wl_all     = (const float*)d_in[2];   // [L,128,128]
    const float* Wr_all     = (const float*)d_in[3];
    const float* att_all    = (const float*)d_in[4];   // [L,8,16]
    const float* bias_all   = (const float*)d_in[5];   // [L,128]
    float* out = (float*)d_out;

    const int N = in_sizes[0] / HID;       // 50000 (multiple of 16)
    const int E = in_sizes[1] / 2;         // 640000
    const int EH = E * HEADS;

    const int* src = edge_index;
    const int* dst = edge_index + E;

    // workspace carve-out
    const size_t nf = (size_t)N * HID;
    float* xl    = (float*)d_ws;
    float* xr    = xl + nf;
    float* bufA  = xr + nf;                 // intermediate node features
    float* mmax  = bufA + nf;
    float* denom = mmax + (size_t)N * HEADS;
    float* lw    = denom + (size_t)N * HEADS;   // [E, H] logits -> weights

    const int TPB = 256;
    const int gemm_blocks = N / 16;                       // 3125
    const int nh_blocks   = (N * HEADS + TPB - 1) / TPB;
    const int nc_blocks   = (N * HID + TPB - 1) / TPB;
    const int eh_blocks   = (EH + TPB - 1) / TPB;

    const float* xin = x;
    for (int l = 0; l < LAYERS; ++l) {
        const float* Wl   = Wl_all   + (size_t)l * HID * HID;
        const float* Wr   = Wr_all   + (size_t)l * HID * HID;
        const float* att  = att_all  + (size_t)l * HEADS * HEAD_DIM;
        const float* bias = bias_all + (size_t)l * HID;
        float* xout = (l == LAYERS - 1) ? out : bufA;     // gemm reads xin first,
                                                          // so xout may alias xin
        gemm_dual_f32<<<gemm_blocks, TPB, 0, stream>>>(xin, Wl, Wr, xl, xr);
        init_md_kernel<<<nh_blocks, TPB, 0, stream>>>(mmax, denom, N * HEADS);
        init_out_kernel<<<nc_blocks, TPB, 0, stream>>>(xout, bias, N * HID);
        edge_logits_kernel<<<eh_blocks, TPB, 0, stream>>>(xl, xr, src, dst, att,
                                                          lw, mmax, EH);
        edge_softmax_kernel<<<eh_blocks, TPB, 0, stream>>>(dst, mmax, lw, denom, EH);
        edge_aggregate_kernel<<<eh_blocks, TPB, 0, stream>>>(src, dst, xl, lw,
                                                             denom, xout, EH);
        xin = xout;
    }
}